// PSLoss_32280974197169
// MI455X (gfx1250) — compile-verified
//
#include <hip/hip_runtime.h>
#include <hip/hip_bf16.h>

#define T_LEN     336
#define B_BATCH   64
#define C_CH      21
#define N_SERIES  (B_BATCH * C_CH)   // 1344
#define K_TOP     3
#define MIN_PATCH 5
#define S_SEG     (T_LEN / MIN_PATCH)  // 67
#define P0_SEG    (K_TOP * S_SEG + K_TOP) // 204
#define P_SEG     (P0_SEG + 1)            // 205 (fallback)
#define NBINS     168                     // rfft bins 1..168
#define FTILES    11                      // 11*16 = 176 >= 168
#define EPS_F     1e-8f
#define MSE_BLOCKS 512

typedef __attribute__((ext_vector_type(2))) float v2f;
typedef __attribute__((ext_vector_type(8))) float v8f;

// ---------------------------------------------------------------------------
// Kernel 1: MSE partial reduction (deterministic tree reduce per block)
// ---------------------------------------------------------------------------
__global__ __launch_bounds__(256)
void ps_mse_kernel(const float* __restrict__ yp, const float* __restrict__ yt,
                   float* __restrict__ partials) {
    __shared__ float sdata[256];
    const int tid = threadIdx.x;
    const int total = B_BATCH * T_LEN * C_CH;
    float acc = 0.0f;
    for (int i = blockIdx.x * blockDim.x + tid; i < total; i += gridDim.x * blockDim.x) {
        float d = yp[i] - yt[i];
        acc += d * d;
    }
    sdata[tid] = acc;
    __syncthreads();
    for (int s = 128; s > 0; s >>= 1) {
        if (tid < s) sdata[tid] += sdata[tid + s];
        __syncthreads();
    }
    if (tid == 0) partials[blockIdx.x] = sdata[0];
}

// ---------------------------------------------------------------------------
// Kernel 2: DFT amplitudes via V_WMMA_F32_16X16X4_F32 + per-series top-3.
// One workgroup per 16-series M-tile; 11 waves, one per 16-bin F-tile.
//
// Twiddles memoized: phase = (f*t) mod 336 indexes a 336-entry {cos,sin}
// LDS table; per K-step the phase is updated incrementally (add + cond-sub),
// so the inner loop is ds_load_b64 + WMMA with no transcendentals.
//
// A (16x4 f32): lane l -> m = l%16, h = l/16; V0 = A[m][2h], V1 = A[m][2h+1]
// B (4x16 f32): lane l -> n = l%16, h = l/16; V0 = B[2h][n], V1 = B[2h+1][n]
// D (16x16 f32): VGPR r, lane l -> row = r + 8*(l/16), col = l%16
// ---------------------------------------------------------------------------
__global__ __launch_bounds__(352)
void ps_dft_top3_kernel(const float* __restrict__ yt, int* __restrict__ top3) {
    __shared__ float  xts[16 * 337];   // padded stride 337 (bank-conflict free)
    __shared__ float2 tw[T_LEN];       // {cos, sin}(2*pi*p/336), p = 0..335
    __shared__ float  amp[16 * 177];   // amp^2 for bins f=1..176 at [m*177 + f-1]

    const int tid   = threadIdx.x;
    const int mtile = blockIdx.x;
    const float w0  = 6.28318530717958647692f / (float)T_LEN;

    // Build twiddle table (336 sincos total per workgroup).
    for (int p = tid; p < T_LEN; p += 352) {
        float s, c;
        __sincosf((float)p * w0, &s, &c);
        tw[p] = make_float2(c, s);
    }

    // Cooperative gather of 16 series x 336 timesteps into LDS.
    for (int idx = tid; idx < 16 * T_LEN; idx += 352) {
        int m = idx / T_LEN;
        int t = idx - m * T_LEN;
        int series = mtile * 16 + m;
        int b = series / C_CH;
        int c = series - b * C_CH;
        xts[m * 337 + t] = yt[(b * T_LEN + t) * C_CH + c];
    }
    __syncthreads();

    const int wave = tid >> 5;
    const int lane = tid & 31;
    const int m    = lane & 15;         // A row / D column index
    const int h    = lane >> 4;         // lane half
    const int f    = 1 + wave * 16 + m; // this lane's frequency bin (1..176)

    v8f accRe = {0.f, 0.f, 0.f, 0.f, 0.f, 0.f, 0.f, 0.f};
    v8f accIm = {0.f, 0.f, 0.f, 0.f, 0.f, 0.f, 0.f, 0.f};

    // Incremental phases: p0 = (f * (kb + 2h)) mod 336, step = (4f) mod 336.
    int p0  = (f * (2 * h)) % T_LEN;
    const int inc = (4 * f) % T_LEN;

    for (int kb = 0; kb < T_LEN; kb += 4) {
        int p1 = p0 + f;                    // f <= 176 < 336 -> one cond-sub
        if (p1 >= T_LEN) p1 -= T_LEN;

        float2 w0v = tw[p0];                // ds_load_b64
        float2 w1v = tw[p1];                // ds_load_b64

        v2f a;
        a.x = xts[m * 337 + kb + 2 * h];    // consecutive -> ds_load_b64
        a.y = xts[m * 337 + kb + 2 * h + 1];

        v2f bc; bc.x = w0v.x; bc.y = w1v.x;
        v2f bs; bs.x = w0v.y; bs.y = w1v.y;

        accRe = __builtin_amdgcn_wmma_f32_16x16x4_f32(
            false, a, false, bc, (short)0, accRe, false, false);
        accIm = __builtin_amdgcn_wmma_f32_16x16x4_f32(
            false, a, false, bs, (short)0, accIm, false, false);

        p0 += inc;                          // inc < 336 -> one cond-sub
        if (p0 >= T_LEN) p0 -= T_LEN;
    }

    // Write amp^2 (re^2 + im^2); bins beyond 168 get sentinel -1.
    for (int r = 0; r < 8; ++r) {
        int mrow = r + 8 * h;
        float re = accRe[r];
        float im = accIm[r];
        float a2 = re * re + im * im;
        if (f > NBINS) a2 = -1.0f;
        amp[mrow * 177 + (f - 1)] = a2;
    }
    __syncthreads();

    // Top-3 scan per series (strict > => lower index wins ties, like lax.top_k)
    if (tid < 16) {
        float b0 = -3.4e38f, b1 = -3.4e38f, b2 = -3.4e38f;
        int   i0 = 0, i1 = 0, i2 = 0;
        for (int j = 0; j < NBINS; ++j) {
            float a = amp[tid * 177 + j];
            if (a > b0)      { b2 = b1; i2 = i1; b1 = b0; i1 = i0; b0 = a; i0 = j; }
            else if (a > b1) { b2 = b1; i2 = i1; b1 = a;  i1 = j; }
            else if (a > b2) { b2 = a;  i2 = j; }
        }
        int series = mtile * 16 + tid;
        top3[series * 3 + 0] = i0 + 1;   // frequency = rel_idx + 1
        top3[series * 3 + 1] = i1 + 1;
        top3[series * 3 + 2] = i2 + 1;
    }
}

// ---------------------------------------------------------------------------
// Kernel 3: per-series segment statistics via LDS prefix sums.
// One workgroup (256 threads) per series.
// ---------------------------------------------------------------------------
__global__ __launch_bounds__(256)
void ps_seg_kernel(const float* __restrict__ yp, const float* __restrict__ ytr,
                   const int* __restrict__ top3,
                   float* __restrict__ structP, float* __restrict__ countP) {
    __shared__ float xt[T_LEN];
    __shared__ float xp[T_LEN];
    __shared__ float pref[5][T_LEN + 1];
    __shared__ int   st[P_SEG];
    __shared__ int   en[P_SEG];
    __shared__ int   code[P0_SEG];
    __shared__ unsigned char val0[P0_SEG];
    __shared__ unsigned char val1[P_SEG];
    __shared__ int   periods[3];
    __shared__ float red[256];
    __shared__ float redc[256];

    const int tid = threadIdx.x;
    const int series = blockIdx.x;
    const int b = series / C_CH;
    const int c = series - b * C_CH;

    for (int t = tid; t < T_LEN; t += 256) {
        int g = (b * T_LEN + t) * C_CH + c;
        xt[t] = ytr[g];
        xp[t] = yp[g];
    }
    if (tid < 3) {
        int f = top3[series * 3 + tid];
        periods[tid] = T_LEN / f;      // floor(T/freq)
    }
    __syncthreads();

    // Prefix sums of xt, xp, xt^2, xp^2, xt*xp (serial per feature: 336 adds)
    if (tid < 5) {
        float run = 0.0f;
        pref[tid][0] = 0.0f;
        for (int t = 0; t < T_LEN; ++t) {
            float a = xt[t], p = xp[t];
            float v;
            switch (tid) {
                case 0: v = a;     break;
                case 1: v = p;     break;
                case 2: v = a * a; break;
                case 3: v = p * p; break;
                default: v = a * p; break;
            }
            run += v;
            pref[tid][t + 1] = run;
        }
    }

    // Build 204 candidate segments (k-major: k*67+s, then 3 remainders)
    if (tid < P0_SEG) {
        int s_, e_, ok;
        if (tid < K_TOP * S_SEG) {
            int k = tid / S_SEG;
            int s = tid - k * S_SEG;
            int p = periods[k];
            s_ = s * p;
            e_ = s_ + p;
            ok = (p >= MIN_PATCH) && (e_ <= T_LEN);
        } else {
            int k = tid - K_TOP * S_SEG;
            int p = periods[k];
            int nseg = T_LEN / p;
            s_ = nseg * p;
            e_ = T_LEN;
            ok = (p >= MIN_PATCH) && ((T_LEN - s_) >= MIN_PATCH);
        }
        st[tid] = s_;
        en[tid] = e_;
        code[tid] = s_ * (T_LEN + 1) + e_;
        val0[tid] = (unsigned char)ok;
    }
    if (tid == P0_SEG) { st[P0_SEG] = 0; en[P0_SEG] = T_LEN; }
    __syncthreads();

    // Dedup against *pre-dedup* validity of earlier segments (reference rule)
    if (tid < P0_SEG) {
        int dup = 0;
        int myc = code[tid];
        for (int j = 0; j < tid; ++j) {
            if (code[j] == myc && val0[j]) { dup = 1; break; }
        }
        val1[tid] = (unsigned char)(val0[tid] && !dup);
    }
    __syncthreads();
    if (tid == 0) {
        int any = 0;
        for (int j = 0; j < P0_SEG; ++j) any |= val1[j];
        val1[P0_SEG] = (unsigned char)(!any);   // fallback [0,T]
    }
    __syncthreads();

    float contrib = 0.0f, cnt = 0.0f;
    if (tid < P_SEG && val1[tid]) {
        int s_ = st[tid], e_ = en[tid];
        float L  = (float)(e_ - s_);
        float s0 = pref[0][e_] - pref[0][s_];
        float s1 = pref[1][e_] - pref[1][s_];
        float s2 = pref[2][e_] - pref[2][s_];
        float s3 = pref[3][e_] - pref[3][s_];
        float s4 = pref[4][e_] - pref[4][s_];
        float mt = s0 / L, mp = s1 / L;
        float vt = fmaxf(s2 / L - mt * mt, 0.0f);
        float vp = fmaxf(s3 / L - mp * mp, 0.0f);
        float cov = s4 / L - mt * mp;
        float denom = sqrtf(L * vp + EPS_F) * sqrtf(L * vt + EPS_F);
        float corr = fminf(fmaxf(L * cov / denom, -1.0f), 1.0f);
        float dc = 1.0f - corr;
        float dv = vp - vt;
        float dm = mp - mt;
        contrib = dc * dc + dv * dv + dm * dm;  // W_CORR=W_VAR=W_MEAN=1
        cnt = 1.0f;
    }
    red[tid]  = contrib;
    redc[tid] = cnt;
    __syncthreads();
    for (int s = 128; s > 0; s >>= 1) {
        if (tid < s) { red[tid] += red[tid + s]; redc[tid] += redc[tid + s]; }
        __syncthreads();
    }
    if (tid == 0) {
        structP[series] = red[0];
        countP[series]  = redc[0];
    }
}

// ---------------------------------------------------------------------------
// Kernel 4: deterministic final combine
// ---------------------------------------------------------------------------
__global__ void ps_final_kernel(const float* __restrict__ mseP,
                                const float* __restrict__ structP,
                                const float* __restrict__ countP,
                                float* __restrict__ out) {
    if (threadIdx.x == 0 && blockIdx.x == 0) {
        float mse = 0.0f;
        for (int i = 0; i < MSE_BLOCKS; ++i) mse += mseP[i];
        mse /= (float)(B_BATCH * T_LEN * C_CH);
        float ss = 0.0f, cc = 0.0f;
        for (int i = 0; i < N_SERIES; ++i) { ss += structP[i]; cc += countP[i]; }
        float n = fmaxf(cc, 1.0f);
        out[0] = 0.5f * mse + 0.5f * (ss / n);
    }
}

// ---------------------------------------------------------------------------
// Launch
// ---------------------------------------------------------------------------
extern "C" void kernel_launch(void* const* d_in, const int* in_sizes, int n_in,
                              void* d_out, int out_size, void* d_ws, size_t ws_size,
                              hipStream_t stream) {
    const float* yp = (const float*)d_in[0];  // y_pred_batch (B,T,C)
    const float* yt = (const float*)d_in[1];  // y_true_batch (B,T,C)

    float* ws      = (float*)d_ws;
    float* mseP    = ws;                         // [0, 512)
    float* structP = ws + MSE_BLOCKS;            // [512, 1856)
    float* countP  = ws + MSE_BLOCKS + N_SERIES; // [1856, 3200)
    int*   top3    = (int*)(ws + MSE_BLOCKS + 2 * N_SERIES); // 1344*3 ints

    ps_mse_kernel<<<MSE_BLOCKS, 256, 0, stream>>>(yp, yt, mseP);
    ps_dft_top3_kernel<<<N_SERIES / 16, 352, 0, stream>>>(yt, top3);
    ps_seg_kernel<<<N_SERIES, 256, 0, stream>>>(yp, yt, top3, structP, countP);
    ps_final_kernel<<<1, 32, 0, stream>>>(mseP, structP, countP, (float*)d_out);
}